// FrameVQVAE_36421322670573
// MI455X (gfx1250) — compile-verified
//
#include <hip/hip_runtime.h>
#include <hip/hip_bf16.h>

typedef __attribute__((ext_vector_type(2))) float v2f;
typedef __attribute__((ext_vector_type(8))) float v8f;

#define B_    1024
#define D_    512
#define E_    16
#define HID_  64
#define K_    512
#define N_    (B_ * D_)
#define BETA_ 0.25f

// im2col K-index (k = cin*3 + dk) -> LDS offset into haloed [cin][D_+2] rows.
// Only ever called with compile-time k (after unroll), so it folds to a const.
__device__ __forceinline__ int koff(int k) {
    return (k / 3) * (D_ + 2) + (k % 3);
}

// ---------------------------------------------------------------------------
// workspace init (harness poisons ws; zero what we accumulate into)
// ---------------------------------------------------------------------------
__global__ void vq_init_kernel(unsigned int* __restrict__ counts,
                               float* __restrict__ loss_acc) {
    int t = threadIdx.x;
    if (t < K_) counts[t] = 0u;
    if (t == 0) *loss_acc = 0.0f;
}

// ---------------------------------------------------------------------------
// encoder: x[B,D] -> z_e_map[B,E,D]
//   conv1 (1->64, relu): VALU, channel-chunked into LDS
//   conv2 (64->16): WMMA GEMM  M=16(E) x K=192(64c*3) x N=512, K-chunked by 48
// A = enc_w2 flat (e*192 + k) ; B = im2col of h via haloed LDS rows.
// ---------------------------------------------------------------------------
__global__ __launch_bounds__(256) void encoder_kernel(
    const float* __restrict__ x,  const float* __restrict__ w1,
    const float* __restrict__ b1, const float* __restrict__ w2,
    const float* __restrict__ b2, float* __restrict__ ze) {
    __shared__ float x_s[D_ + 2];              // zero halo
    __shared__ float h_s[16 * (D_ + 2)];       // 16-channel chunk, haloed
    __shared__ float w2_s[E_ * HID_ * 3];      // A matrix: e*192 + k
    __shared__ float w1_s[HID_ * 3];
    __shared__ float b1_s[HID_];
    __shared__ float b2_s[E_];

    const int t = threadIdx.x;
    const int b = blockIdx.x;

    x_s[t + 1]   = x[b * D_ + t];
    x_s[t + 257] = x[b * D_ + t + 256];
    if (t == 0) { x_s[0] = 0.0f; x_s[D_ + 1] = 0.0f; }
    if (t < 16) { h_s[t * (D_ + 2)] = 0.0f; h_s[t * (D_ + 2) + D_ + 1] = 0.0f; }
    for (int i = t; i < E_ * HID_ * 3; i += 256) w2_s[i] = w2[i];
    if (t < HID_ * 3) w1_s[t] = w1[t];
    if (t < HID_)     b1_s[t] = b1[t];
    if (t < E_)       b2_s[t] = b2[t];

    const int wave = t >> 5;
    const int lane = t & 31;
    const int hh   = lane >> 4;
    const int lm   = lane & 15;

    __syncthreads();

    // accumulators: 4 N-tiles per wave; C row M = 8*hh + j  (= output e)
    v8f acc[4];
#pragma unroll
    for (int nt = 0; nt < 4; ++nt)
#pragma unroll
        for (int j = 0; j < 8; ++j) acc[nt][j] = b2_s[8 * hh + j];

    for (int cc = 0; cc < 4; ++cc) {           // 4 chunks of 16 hidden channels
        // ---- conv1 chunk (VALU, cooperative across block) ----
#pragma unroll
        for (int ci = 0; ci < 16; ++ci) {
            int c = cc * 16 + ci;
            float wa = w1_s[c * 3 + 0], wb = w1_s[c * 3 + 1],
                  wc = w1_s[c * 3 + 2], bias = b1_s[c];
#pragma unroll
            for (int p = 0; p < 2; ++p) {
                int l = t + p * 256;
                float v = bias + wa * x_s[l] + wb * x_s[l + 1] + wc * x_s[l + 2];
                h_s[ci * (D_ + 2) + l + 1] = fmaxf(v, 0.0f);
            }
        }
        __syncthreads();
        // ---- conv2 K-slice via WMMA: local K = 48 -> 12 chunks of 4 ----
#pragma unroll
        for (int q = 0; q < 12; ++q) {
            int kl = 4 * q + 2 * hh;           // this lane's K pair (even)
            v2f a = *(const v2f*)&w2_s[lm * 192 + cc * 48 + kl];
            int offx = hh ? koff(4 * q + 2) : koff(4 * q);
            int offy = hh ? koff(4 * q + 3) : koff(4 * q + 1);
#pragma unroll
            for (int nt = 0; nt < 4; ++nt) {
                int l0 = (wave * 4 + nt) * 16;
                v2f bb;
                bb.x = h_s[offx + l0 + lm];
                bb.y = h_s[offy + l0 + lm];
                acc[nt] = __builtin_amdgcn_wmma_f32_16x16x4_f32(
                              false, a, false, bb, (short)0, acc[nt],
                              false, false);
            }
        }
        __syncthreads();
    }
#pragma unroll
    for (int nt = 0; nt < 4; ++nt) {
        int l0 = (wave * 4 + nt) * 16;
#pragma unroll
        for (int j = 0; j < 8; ++j)
            ze[(b * E_ + 8 * hh + j) * D_ + l0 + lm] = acc[nt][j];
    }
}

// ---------------------------------------------------------------------------
// VQ: argmin_n ||c_n||^2 - 2 z.c_n via V_WMMA_F32_16X16X4_F32.
// One wave = one 16-row tile vs all 512 codes (32 N-tiles x 4 K-chunks).
// ---------------------------------------------------------------------------
__global__ __launch_bounds__(256) void vq_kernel(
    const float* __restrict__ ze, const float* __restrict__ cb,
    float* __restrict__ zq, float* __restrict__ idx_out,
    unsigned int* __restrict__ counts) {
    __shared__ float cb_s[K_ * E_];
    __shared__ float nrm_s[K_];

    const int t = threadIdx.x;

    __builtin_prefetch(cb + t * 32, 0, 1);     // global_prefetch of codebook
    for (int i = t; i < K_ * E_; i += 256) cb_s[i] = cb[i];
    __syncthreads();
    for (int n = t; n < K_; n += 256) {
        float s = 0.0f;
#pragma unroll
        for (int e = 0; e < E_; ++e) { float v = cb_s[n * E_ + e]; s += v * v; }
        nrm_s[n] = s;
    }
    __syncthreads();

    const int wave = t >> 5;
    const int lane = t & 31;
    const int hh   = lane >> 4;
    const int lm   = lane & 15;
    const int m0   = (blockIdx.x * 8 + wave) * 16;
    const int b    = m0 / D_;
    const int l0   = m0 % D_;

    // A = -2 * z   (distance term folds into the WMMA accumulate)
    v2f a[4];
#pragma unroll
    for (int q = 0; q < 4; ++q) {
        int k0 = 4 * q + 2 * hh;
        a[q].x = -2.0f * ze[(b * E_ + k0)     * D_ + (l0 + lm)];
        a[q].y = -2.0f * ze[(b * E_ + k0 + 1) * D_ + (l0 + lm)];
    }

    float minv[8]; int mini[8];
#pragma unroll
    for (int j = 0; j < 8; ++j) { minv[j] = 3.4e38f; mini[j] = 0; }

    for (int tile = 0; tile < K_ / 16; ++tile) {
        int n = tile * 16 + lm;
        float nn = nrm_s[n];
        v8f c;
#pragma unroll
        for (int j = 0; j < 8; ++j) c[j] = nn;  // C init = ||c_n||^2
#pragma unroll
        for (int q = 0; q < 4; ++q) {
            int k0 = 4 * q + 2 * hh;
            v2f bb;
            bb.x = cb_s[n * E_ + k0];
            bb.y = cb_s[n * E_ + k0 + 1];
            c = __builtin_amdgcn_wmma_f32_16x16x4_f32(
                    false, a[q], false, bb, (short)0, c, false, false);
        }
#pragma unroll
        for (int j = 0; j < 8; ++j)
            if (c[j] < minv[j]) { minv[j] = c[j]; mini[j] = n; }
    }

    // min-reduce across the 16 lanes of each half
#pragma unroll
    for (int j = 0; j < 8; ++j) {
#pragma unroll
        for (int m = 8; m >= 1; m >>= 1) {
            float ov = __shfl_xor(minv[j], m, 32);
            int   oi = __shfl_xor(mini[j], m, 32);
            if (ov < minv[j] || (ov == minv[j] && oi < mini[j])) {
                minv[j] = ov; mini[j] = oi;
            }
        }
    }
    int idx_r[16];
#pragma unroll
    for (int j = 0; j < 8; ++j) {
        idx_r[j]     = __shfl(mini[j], 0, 32);
        idx_r[8 + j] = __shfl(mini[j], 16, 32);
    }

    if (lane == 0) {
#pragma unroll
        for (int r = 0; r < 16; ++r) {
            idx_out[m0 + r] = (float)idx_r[r];
            atomicAdd(&counts[idx_r[r]], 1u);
        }
    }
#pragma unroll
    for (int rr = 0; rr < 8; ++rr) {
        int row = 2 * rr + hh;
        int u   = hh ? idx_r[2 * rr + 1] : idx_r[2 * rr];
        zq[(b * E_ + lm) * D_ + (l0 + row)] = cb_s[u * E_ + lm];
    }
}

// ---------------------------------------------------------------------------
// decoder: z_q_map[B,E,D] -> x_hat[B,D]; also accumulates sum((z_q-z_e)^2)
//   conv1 (16->64, relu): WMMA GEMM  M=64 x K=48(16e*3) x N=512,
//                         M-chunked by 16 (4 chunks), results staged in LDS
//   conv2 (64->1): VALU accumulate per chunk
// Dynamic LDS (~79 KB) to stay clear of any static-LDS cap.
// ---------------------------------------------------------------------------
__global__ __launch_bounds__(256) void decoder_kernel(
    const float* __restrict__ zq, const float* __restrict__ ze,
    const float* __restrict__ w1, const float* __restrict__ b1,
    const float* __restrict__ w2, const float* __restrict__ b2,
    float* __restrict__ xhat, float* __restrict__ loss_acc) {
    extern __shared__ float dsm[];
    float* zq_s = dsm;                          // [16][D_+2] haloed
    float* h_s  = zq_s + E_ * (D_ + 2);         // [16][D_+2] haloed (chunk)
    float* w1_s = h_s + 16 * (D_ + 2);          // A matrix: c*48 + k
    float* b1_s = w1_s + HID_ * E_ * 3;         // [64]
    float* red  = b1_s + HID_;                  // [256]

    const int t = threadIdx.x;
    const int b = blockIdx.x;

    float lsum = 0.0f;
#pragma unroll
    for (int e = 0; e < E_; ++e) {
#pragma unroll
        for (int p = 0; p < 2; ++p) {
            int l = t + p * 256;
            float q  = zq[(b * E_ + e) * D_ + l];
            float zv = ze[(b * E_ + e) * D_ + l];
            float d  = q - zv;
            lsum += d * d;
            zq_s[e * (D_ + 2) + l + 1] = q;
        }
    }
    if (t < E_) { zq_s[t * (D_ + 2)] = 0.0f; zq_s[t * (D_ + 2) + D_ + 1] = 0.0f; }
    if (t < 16) { h_s[t * (D_ + 2)]  = 0.0f; h_s[t * (D_ + 2) + D_ + 1]  = 0.0f; }
    for (int i = t; i < HID_ * E_ * 3; i += 256) w1_s[i] = w1[i];
    if (t < HID_) b1_s[t] = b1[t];

    const int wave = t >> 5;
    const int lane = t & 31;
    const int hh   = lane >> 4;
    const int lm   = lane & 15;

    float xacc0 = b2[0], xacc1 = b2[0];
    __syncthreads();

    for (int mt = 0; mt < 4; ++mt) {            // 4 chunks of 16 out-channels
        // ---- conv1 chunk via WMMA: K = 48 -> 12 chunks of 4 ----
        v8f acc[4];
#pragma unroll
        for (int nt = 0; nt < 4; ++nt)
#pragma unroll
            for (int j = 0; j < 8; ++j)
                acc[nt][j] = b1_s[mt * 16 + 8 * hh + j];
#pragma unroll
        for (int q = 0; q < 12; ++q) {
            int kl = 4 * q + 2 * hh;
            v2f a = *(const v2f*)&w1_s[(mt * 16 + lm) * 48 + kl];
            int offx = hh ? koff(4 * q + 2) : koff(4 * q);
            int offy = hh ? koff(4 * q + 3) : koff(4 * q + 1);
#pragma unroll
            for (int nt = 0; nt < 4; ++nt) {
                int l0 = (wave * 4 + nt) * 16;
                v2f bb;
                bb.x = zq_s[offx + l0 + lm];
                bb.y = zq_s[offy + l0 + lm];
                acc[nt] = __builtin_amdgcn_wmma_f32_16x16x4_f32(
                              false, a, false, bb, (short)0, acc[nt],
                              false, false);
            }
        }
#pragma unroll
        for (int nt = 0; nt < 4; ++nt) {
            int l0 = (wave * 4 + nt) * 16;
#pragma unroll
            for (int j = 0; j < 8; ++j)
                h_s[(8 * hh + j) * (D_ + 2) + l0 + lm + 1] =
                    fmaxf(acc[nt][j], 0.0f);
        }
        __syncthreads();
        // ---- conv2 accumulate (64->1) over this chunk ----
#pragma unroll
        for (int ci = 0; ci < 16; ++ci) {
            int c = mt * 16 + ci;
            float k0 = w2[c * 3 + 0], k1 = w2[c * 3 + 1], k2 = w2[c * 3 + 2];
            const float* hr = &h_s[ci * (D_ + 2)];
            xacc0 += k0 * hr[t]       + k1 * hr[t + 1]   + k2 * hr[t + 2];
            xacc1 += k0 * hr[t + 256] + k1 * hr[t + 257] + k2 * hr[t + 258];
        }
        __syncthreads();
    }
    xhat[b * D_ + t]       = xacc0;
    xhat[b * D_ + t + 256] = xacc1;

    red[t] = lsum;
    __syncthreads();
    for (int s = 128; s > 0; s >>= 1) {
        if (t < s) red[t] += red[t + s];
        __syncthreads();
    }
    if (t == 0) atomicAdd(loss_acc, red[0]);
}

// ---------------------------------------------------------------------------
// finalize: quant_loss = 1.25 * mean((zq-ze)^2); perplexity from histogram
// ---------------------------------------------------------------------------
__global__ void finalize_kernel(const unsigned int* __restrict__ counts,
                                const float* __restrict__ loss_acc,
                                float* __restrict__ quant_out,
                                float* __restrict__ perp_out) {
    __shared__ float red[512];
    int t = threadIdx.x;
    float p = (float)counts[t] / (float)N_;
    red[t] = p * logf(p + 1e-10f);
    __syncthreads();
    for (int s = 256; s > 0; s >>= 1) {
        if (t < s) red[t] += red[t + s];
        __syncthreads();
    }
    if (t == 0) {
        *perp_out  = expf(-red[0]);
        *quant_out = (1.0f + BETA_) * (*loss_acc) / ((float)N_ * (float)E_);
    }
}

// ---------------------------------------------------------------------------
extern "C" void kernel_launch(void* const* d_in, const int* in_sizes, int n_in,
                              void* d_out, int out_size, void* d_ws, size_t ws_size,
                              hipStream_t stream) {
    const float* x   = (const float*)d_in[0];
    const float* ew1 = (const float*)d_in[1];
    const float* eb1 = (const float*)d_in[2];
    const float* ew2 = (const float*)d_in[3];
    const float* eb2 = (const float*)d_in[4];
    const float* dw1 = (const float*)d_in[5];
    const float* db1 = (const float*)d_in[6];
    const float* dw2 = (const float*)d_in[7];
    const float* db2 = (const float*)d_in[8];
    const float* cb  = (const float*)d_in[9];

    float* out    = (float*)d_out;
    float* xhat   = out;                        // [B,D]
    float* ze     = xhat + N_;                  // [B,E,D]
    float* zq     = ze + (size_t)N_ * E_;       // [B,E,D]
    float* idxf   = zq + (size_t)N_ * E_;       // [B*D]
    float* qloss  = idxf + N_;                  // [1]
    float* perp   = qloss + 1;                  // [1]

    unsigned int* counts = (unsigned int*)d_ws;             // 512 u32
    float* loss_acc = (float*)((char*)d_ws + K_ * sizeof(unsigned int));

    const size_t dec_lds =
        (size_t)(2 * E_ * (D_ + 2) + HID_ * E_ * 3 + HID_ + 256) * sizeof(float);

    vq_init_kernel<<<1, 512, 0, stream>>>(counts, loss_acc);
    encoder_kernel<<<B_, 256, 0, stream>>>(x, ew1, eb1, ew2, eb2, ze);
    vq_kernel<<<N_ / (16 * 8), 256, 0, stream>>>(ze, cb, zq, idxf, counts);
    decoder_kernel<<<B_, 256, 0, stream>>>(zq, ze, dw1, db1, dw2, db2,
                                           xhat, loss_acc);
    finalize_kernel<<<1, 512, 0, stream>>>(counts, loss_acc, qloss, perp);
}